// Encoder_12850542150111
// MI455X (gfx1250) — compile-verified
//
#include <hip/hip_runtime.h>

typedef float v2f __attribute__((ext_vector_type(2)));
typedef float v8f __attribute__((ext_vector_type(8)));

#define K_DIM 256
#define LDS_PITCH 260   // 260 % 64 == 4 -> conflict-free strided A-fragment reads

// ---------------------------------------------------------------------------
// GEMM: Hout[N x OUT] = (RELU_IN ? relu(X) : X)[N x 256] @ W[256 x OUT]
// WT is the transposed weight: WT[col * 256 + k] = W[k * OUT + col], so each
// B fragment (K-pair for one column) is one contiguous global_load_b64.
// One block = 16 rows, 8 waves, each wave owns CT 16-wide column tiles.
// Uses V_WMMA_F32_16X16X4_F32 (fp32 in, fp32 accumulate).
// ---------------------------------------------------------------------------
template <int CT, bool RELU_IN>
__global__ __launch_bounds__(256) void gemm_wmma_kernel(
    const float* __restrict__ X, const float* __restrict__ WT,
    float* __restrict__ Hout, int OUT) {
  __shared__ float xs[16 * LDS_PITCH];

  const int row0 = blockIdx.x * 16;
  const int t = threadIdx.x;

  // Cooperative load: 16 rows x 256 cols fp32 into LDS (float4 moves).
  {
    const int r  = t >> 4;          // 0..15  row of tile
    const int c0 = (t & 15) * 16;   // 0..240 start col
    const float4* xg = (const float4*)(X + (size_t)(row0 + r) * K_DIM + c0);
    float4* xl = (float4*)(&xs[r * LDS_PITCH + c0]);
#pragma unroll
    for (int i = 0; i < 4; ++i) {
      float4 v = xg[i];
      if (RELU_IN) {
        v.x = fmaxf(v.x, 0.f); v.y = fmaxf(v.y, 0.f);
        v.z = fmaxf(v.z, 0.f); v.w = fmaxf(v.w, 0.f);
      }
      xl[i] = v;
    }
  }
  __syncthreads();

  const int wave = t >> 5;
  const int lane = t & 31;
  const int half = lane >> 4;   // A/B: half=0 -> K pair (0,1); half=1 -> (2,3)
  const int m    = lane & 15;   // A row within tile / B-C column within tile

  v8f zero = {};
  v8f acc[CT];
  const float* wcol[CT];
#pragma unroll
  for (int c = 0; c < CT; ++c) {
    acc[c] = zero;
    wcol[c] = WT + (size_t)((wave * CT + c) * 16 + m) * K_DIM + 2 * half;
  }

  const float* xrow = &xs[m * LDS_PITCH + 2 * half];

#pragma unroll 4
  for (int k0 = 0; k0 < K_DIM; k0 += 4) {
    // A fragment (ISA 32-bit A 16x4): V0 = K[k0+2*half], V1 = K[k0+2*half+1]
    v2f a = *(const v2f*)(xrow + k0);
#pragma unroll
    for (int c = 0; c < CT; ++c) {
      // B fragment: contiguous K-pair from transposed weights (one b64 load)
      v2f b = *(const v2f*)(wcol[c] + k0);
      acc[c] = __builtin_amdgcn_wmma_f32_16x16x4_f32(
          false, a, false, b, (short)0, acc[c], false, false);
    }
  }

  // C/D layout: VGPR r holds M = r + 8*half, N = lane&15
#pragma unroll
  for (int c = 0; c < CT; ++c) {
    const int col = (wave * CT + c) * 16 + m;
#pragma unroll
    for (int r = 0; r < 8; ++r) {
      Hout[(size_t)(row0 + 8 * half + r) * OUT + col] = acc[c][r];
    }
  }
}

// ---------------------------------------------------------------------------
// Weight transpose: WT[col * K + k] = W[k * OUT + col]   (tiny, done once)
// ---------------------------------------------------------------------------
__global__ void transpose_kernel(const float* __restrict__ W,
                                 float* __restrict__ WT, int K, int OUT) {
  int idx = blockIdx.x * 256 + threadIdx.x;
  if (idx >= K * OUT) return;
  int col = idx / K;
  int k = idx - col * K;
  WT[idx] = W[(size_t)k * OUT + col];
}

// ---------------------------------------------------------------------------
// Graph-side kernels
// ---------------------------------------------------------------------------
__global__ void fill_kernel(float* p, float v, int n) {
  int i = blockIdx.x * 256 + threadIdx.x;
  if (i < n) p[i] = v;
}

__global__ void deg_kernel(const int* __restrict__ dst, float* deg, int E) {
  int i = blockIdx.x * 256 + threadIdx.x;
  if (i < E) unsafeAtomicAdd(&deg[dst[i]], 1.0f);
}

__global__ void rsqrt_kernel(float* d, int n) {
  int i = blockIdx.x * 256 + threadIdx.x;
  if (i < n) {
    float v = d[i];
    d[i] = v > 0.f ? rsqrtf(v) : 0.f;
  }
}

// out[i][f] = b[f] + dinv[i]^2 * h[i][f]   (bias + self-loop term), float4-wide
template <int F>
__global__ void agg_init_kernel(float4* __restrict__ out,
                                const float4* __restrict__ h,
                                const float4* __restrict__ b4,
                                const float* __restrict__ dinv, int N) {
  int idx = blockIdx.x * 256 + threadIdx.x;       // over N * F/4
  if (idx >= N * (F / 4)) return;
  int i = idx / (F / 4);
  int f4 = idx & (F / 4 - 1);
  float di = dinv[i];
  float s = di * di;
  float4 hv = h[idx];
  float4 bv = b4[f4];
  float4 o;
  o.x = bv.x + s * hv.x;
  o.y = bv.y + s * hv.y;
  o.z = bv.z + s * hv.z;
  o.w = bv.w + s * hv.w;
  out[idx] = o;
}

// One wave per edge: out[dst] += dinv[src]*dinv[dst] * h[src]
// float4 gathers (global_load_b128) + scalar f32 HW atomics into L2.
template <int F>
__global__ void edge_scatter_kernel(const int* __restrict__ src,
                                    const int* __restrict__ dst,
                                    const float* __restrict__ dinv,
                                    const float* __restrict__ h,
                                    float* __restrict__ out, int E) {
  int gid = blockIdx.x * 256 + threadIdx.x;
  int e = gid >> 5;
  int lane = gid & 31;
  if (e >= E) return;
  int s = src[e];
  int d = dst[e];
  float nrm = dinv[s] * dinv[d];
  const float4* hs4 = (const float4*)(h + (size_t)s * F);
  float* od = out + (size_t)d * F;
#pragma unroll
  for (int i = 0; i < F / 128; ++i) {
    int q = lane + i * 32;
    float4 v = hs4[q];
    int f = q * 4;
    unsafeAtomicAdd(&od[f + 0], nrm * v.x);
    unsafeAtomicAdd(&od[f + 1], nrm * v.y);
    unsafeAtomicAdd(&od[f + 2], nrm * v.z);
    unsafeAtomicAdd(&od[f + 3], nrm * v.w);
  }
}

__global__ void relu4_kernel(float4* p, int n4) {
  int i = blockIdx.x * 256 + threadIdx.x;
  if (i < n4) {
    float4 v = p[i];
    v.x = fmaxf(v.x, 0.f); v.y = fmaxf(v.y, 0.f);
    v.z = fmaxf(v.z, 0.f); v.w = fmaxf(v.w, 0.f);
    p[i] = v;
  }
}

// ---------------------------------------------------------------------------
extern "C" void kernel_launch(void* const* d_in, const int* in_sizes, int n_in,
                              void* d_out, int out_size, void* d_ws, size_t ws_size,
                              hipStream_t stream) {
  const float* x  = (const float*)d_in[0];
  const int*   ei = (const int*)d_in[1];
  const float* W1 = (const float*)d_in[2];
  const float* b1 = (const float*)d_in[3];
  const float* W2 = (const float*)d_in[4];
  const float* b2 = (const float*)d_in[5];
  float* out = (float*)d_out;

  const int IN  = 256;
  const int N   = in_sizes[0] / IN;   // 100000 (divisible by 16)
  const int E   = in_sizes[1] / 2;    // 1,600,000
  const int H   = in_sizes[2] / IN;   // 256
  const int OUT = in_sizes[4] / H;    // 128
  const int* srcp = ei;
  const int* dstp = ei + E;

  char* ws = (char*)d_ws;
  auto align_up = [](size_t v) { return (v + 255) & ~(size_t)255; };
  size_t off = 0;
  float* dinv = (float*)(ws + off); off += align_up((size_t)N * 4);
  float* wt1  = (float*)(ws + off); off += align_up((size_t)IN * H * 4);
  float* wt2  = (float*)(ws + off); off += align_up((size_t)H * OUT * 4);
  float* hbuf = (float*)(ws + off); off += align_up((size_t)N * H * 4);
  float* y1   = (float*)(ws + off);

  const int TB = 256;

  // Transposed weights (contiguous K-pairs for WMMA B fragments)
  transpose_kernel<<<(IN * H + TB - 1) / TB, TB, 0, stream>>>(W1, wt1, IN, H);
  transpose_kernel<<<(H * OUT + TB - 1) / TB, TB, 0, stream>>>(W2, wt2, H, OUT);

  // Degrees (deg = 1 for self-loop, + count of incoming edges) -> dinv in place
  fill_kernel<<<(N + TB - 1) / TB, TB, 0, stream>>>(dinv, 1.0f, N);
  deg_kernel<<<(E + TB - 1) / TB, TB, 0, stream>>>(dstp, dinv, E);
  rsqrt_kernel<<<(N + TB - 1) / TB, TB, 0, stream>>>(dinv, N);

  // ---- Layer 1: h1 = x @ W1 ; y1 = seg_sum(norm * h1[src]) + b1 ----
  gemm_wmma_kernel<2, false><<<N / 16, TB, 0, stream>>>(x, wt1, hbuf, H);
  agg_init_kernel<256><<<(N * 64 + TB - 1) / TB, TB, 0, stream>>>(
      (float4*)y1, (const float4*)hbuf, (const float4*)b1, dinv, N);
  edge_scatter_kernel<256><<<((size_t)E * 32 + TB - 1) / TB, TB, 0, stream>>>(
      srcp, dstp, dinv, hbuf, y1, E);

  // ---- Layer 2: h2 = relu(y1) @ W2 (reuse hbuf); out = agg + b2, relu ----
  gemm_wmma_kernel<1, true><<<N / 16, TB, 0, stream>>>(y1, wt2, hbuf, OUT);
  agg_init_kernel<128><<<(N * 32 + TB - 1) / TB, TB, 0, stream>>>(
      (float4*)out, (const float4*)hbuf, (const float4*)b2, dinv, N);
  edge_scatter_kernel<128><<<((size_t)E * 32 + TB - 1) / TB, TB, 0, stream>>>(
      srcp, dstp, dinv, hbuf, out, E);
  relu4_kernel<<<(N * OUT / 4 + TB - 1) / TB, TB, 0, stream>>>(
      (float4*)out, N * OUT / 4);
}